// NodeDecoder_90941637526230
// MI455X (gfx1250) — compile-verified
//
#include <hip/hip_runtime.h>
#include <math.h>

// Problem dims (fixed by reference)
constexpr int Bn  = 256;   // batch
constexpr int Vn  = 16;    // vehicles
constexpr int Nn  = 1024;  // nodes
constexpr int Dn  = 512;   // feature dim
constexpr int K3D = 3 * Dn; // 1536

typedef __attribute__((ext_vector_type(2))) float v2f;
typedef __attribute__((ext_vector_type(4))) float v4f;
typedef __attribute__((ext_vector_type(8))) float v8f;

// ---------------------------------------------------------------------------
// Phase 0: build cat = [g_node | mean_v(Z_veh) | g_graph]   [B, 3D]
// ---------------------------------------------------------------------------
__global__ void build_cat_kernel(const float* __restrict__ g_node,
                                 const float* __restrict__ Z_veh,
                                 const float* __restrict__ g_graph,
                                 float* __restrict__ cat) {
  int idx = blockIdx.x * blockDim.x + threadIdx.x;   // over Bn*K3D
  int b = idx / K3D;
  int j = idx % K3D;
  float v;
  if (j < Dn) {
    v = g_node[b * Dn + j];
  } else if (j < 2 * Dn) {
    int d = j - Dn;
    const float* p = Z_veh + (size_t)b * Vn * Dn + d;
    float s = 0.0f;
#pragma unroll
    for (int t = 0; t < Vn; ++t) s += p[t * Dn];
    v = s * (1.0f / (float)Vn);
  } else {
    v = g_graph[b * Dn + (j - 2 * Dn)];
  }
  cat[idx] = v;
}

// ---------------------------------------------------------------------------
// WMMA f32 GEMM:  C[M,N] = act(A[M,K] @ B' + bias)
//   TRANSB=1: B'(k,n) = Bm[n*LDB + k]   (i.e. A @ Bm^T, torch Linear weight)
//   TRANSB=0: B'(k,n) = Bm[k*LDB + n]   (i.e. A @ Bm)
// One wave computes one 16x16 tile with V_WMMA_F32_16X16X4_F32.
// Launch with exactly (Mtiles*Ntiles)/8 blocks of 256 threads -> EXEC all-1s.
// ---------------------------------------------------------------------------
template <int KDIM, int LDB, bool TRANSB, bool RELU, bool BIAS>
__global__ void gemm_wmma_f32(const float* __restrict__ A,
                              const float* __restrict__ Bm,
                              const float* __restrict__ bias,
                              float* __restrict__ C,
                              int nTilesN) {
  const int lane = threadIdx.x & 31;
  const int wave = threadIdx.x >> 5;
  const int tile = blockIdx.x * (blockDim.x >> 5) + wave;
  const int tM = tile / nTilesN;
  const int tN = tile - tM * nTilesN;
  const int row0 = tM * 16;
  const int col0 = tN * 16;
  const int half = lane >> 4;      // 0: K pair {0,1}, 1: K pair {2,3}
  const int l = lane & 15;         // M index for A, N index for B/C

  v8f acc = {};
  const float* arow = A + (size_t)(row0 + l) * KDIM + 2 * half;

  for (int k0 = 0; k0 < KDIM; k0 += 4) {
    // A 16x4 layout (ISA 7.12.2): lanes 0-15 -> {K0,K1}, lanes 16-31 -> {K2,K3}
    v2f a = *(const v2f*)(arow + k0);
    v2f b;
    if (TRANSB) {
      b = *(const v2f*)(Bm + (size_t)(col0 + l) * LDB + (k0 + 2 * half));
    } else {
      int k = k0 + 2 * half;
      b.x = Bm[(size_t)k * LDB + col0 + l];
      b.y = Bm[(size_t)(k + 1) * LDB + col0 + l];
    }
    // 8 args: (neg_a, A, neg_b, B, c_mod, C, reuse_a, reuse_b)
    acc = __builtin_amdgcn_wmma_f32_16x16x4_f32(false, a, false, b,
                                                (short)0, acc, false, false);
  }

  const int ldc = nTilesN * 16;
  float bv = 0.0f;
  if (BIAS) bv = bias[col0 + l];
#pragma unroll
  for (int r = 0; r < 8; ++r) {
    // C/D layout: VGPR r = row (r + 8*half), N = l
    float x = acc[r] + bv;
    if (RELU) x = fmaxf(x, 0.0f);
    C[(size_t)(row0 + r + 8 * half) * ldc + col0 + l] = x;
  }
}

// ---------------------------------------------------------------------------
// Phase 2 (bandwidth-bound): logits[b,n] = mask ? 10*tanh(dot(Z[b,n,:],qk[b,:])/sqrt(D))
//                                               : -inf
// One wave per row; 32 lanes cover the contiguous 2KB row with b128 NT loads.
// qk[b] staged in LDS. Block = 8 waves = 8 consecutive rows of one batch b.
// ---------------------------------------------------------------------------
__global__ void logits_kernel(const float* __restrict__ Z_node,
                              const float* __restrict__ qk,
                              const unsigned char* __restrict__ mask,
                              float* __restrict__ out) {
  __shared__ float sq[Dn];
  const int blk = blockIdx.x;          // Bn * (Nn/8) blocks
  const int b = blk >> 7;              // Nn/8 == 128
  const int n = ((blk & 127) << 3) + (threadIdx.x >> 5);
  const int lane = threadIdx.x & 31;

  // stage qk[b][0..511] into LDS (2 floats per thread)
  for (int i = threadIdx.x; i < Dn; i += blockDim.x) sq[i] = qk[b * Dn + i];
  __syncthreads();

  const v4f* zp = (const v4f*)(Z_node + ((size_t)b * Nn + n) * Dn);
  float s = 0.0f;
#pragma unroll
  for (int j = 0; j < 4; ++j) {
    int vi = j * 32 + lane;                       // coalesced 512B per wave-load
    v4f z = __builtin_nontemporal_load(zp + vi);  // 512MB stream: don't thrash L2
    const float* q4 = &sq[vi * 4];
    s += z.x * q4[0] + z.y * q4[1] + z.z * q4[2] + z.w * q4[3];
  }
  // wave32 butterfly reduction
#pragma unroll
  for (int off = 16; off; off >>= 1) s += __shfl_xor(s, off, 32);

  if (lane == 0) {
    const float inv_scale = 0.044194173824159216f;  // 1/sqrt(512)
    float lg = 10.0f * tanhf(s * inv_scale);
    size_t o = (size_t)b * Nn + n;
    out[o] = mask[o] ? lg : -INFINITY;
  }
}

// ---------------------------------------------------------------------------
extern "C" void kernel_launch(void* const* d_in, const int* in_sizes, int n_in,
                              void* d_out, int out_size, void* d_ws, size_t ws_size,
                              hipStream_t stream) {
  const float* g_node  = (const float*)d_in[0];
  const float* Z_veh   = (const float*)d_in[1];
  const float* g_graph = (const float*)d_in[2];
  const float* Z_node  = (const float*)d_in[3];
  const unsigned char* mask = (const unsigned char*)d_in[4];  // jax bool = 1 byte
  const float* W_ctx   = (const float*)d_in[5];
  const float* b_ctx   = (const float*)d_in[6];
  const float* Wq      = (const float*)d_in[7];
  const float* Wk      = (const float*)d_in[8];
  float* out = (float*)d_out;

  // Workspace (fully rewritten each call; regions aliased once dead):
  //   [0, 1.5MB)   cat [B,3D]   -> later reused for q  [B,D]
  //   [1.5, 2MB)   ctx [B,D]    -> later reused for qk [B,D]
  float* cat = (float*)d_ws;
  float* ctx = cat + (size_t)Bn * K3D;
  float* q   = cat;   // cat dead after ctx GEMM
  float* qk  = ctx;   // ctx dead after q GEMM

  // Phase 0: concat + vehicle mean
  build_cat_kernel<<<(Bn * K3D) / 256, 256, 0, stream>>>(g_node, Z_veh, g_graph, cat);

  // Phase 1: three small dense GEMMs on the WMMA pipe (512 tiles = 64 blocks each)
  // ctx = relu(cat @ W_ctx^T + b_ctx)      [256,1536]x[1536,512]
  gemm_wmma_f32<K3D, K3D, true, true, true>
      <<<64, 256, 0, stream>>>(cat, W_ctx, b_ctx, ctx, Dn / 16);
  // q = ctx @ Wq^T                          [256,512]x[512,512]
  gemm_wmma_f32<Dn, Dn, true, false, false>
      <<<64, 256, 0, stream>>>(ctx, Wq, nullptr, q, Dn / 16);
  // qk = q @ Wk                             [256,512]x[512,512]
  gemm_wmma_f32<Dn, Dn, false, false, false>
      <<<64, 256, 0, stream>>>(q, Wk, nullptr, qk, Dn / 16);

  // Phase 2: stream Z_node (512MB, HBM-bound), batched row-dot + tanh + mask
  logits_kernel<<<Bn * (Nn / 8), 256, 0, stream>>>(Z_node, qk, mask, out);
}